// SVC_61349312856346
// MI455X (gfx1250) — compile-verified
//
#include <hip/hip_runtime.h>

#define WIRES 14
#define NSTATE (1 << WIRES)        // 16384 amplitudes / sample
#define NPAIRS 13
#define THREADS 256                 // 8 wave32 per workgroup
#define APT (NSTATE / THREADS)      // 64 amplitudes per thread

typedef __attribute__((ext_vector_type(2))) float v2f;
typedef __attribute__((ext_vector_type(8))) float v8f;

// ============================================================================
// Kernel 1: fuse each 10-gate variational block (acting on adjacent qubits
// q1,q1+1) into a single 4x4 complex unitary. 13 blocks, one thread each.
// Output layout in ws: per block k, 32 floats: Ur[4][4] row-major, Ui[4][4].
// ============================================================================

__device__ __forceinline__ void cmul4(const float Gr[4][4], const float Gi[4][4],
                                      float Ur[4][4], float Ui[4][4]) {
  float Rr[4][4], Ri[4][4];
  for (int i = 0; i < 4; ++i)
    for (int j = 0; j < 4; ++j) {
      float ar = 0.f, ai = 0.f;
      for (int k = 0; k < 4; ++k) {
        ar += Gr[i][k] * Ur[k][j] - Gi[i][k] * Ui[k][j];
        ai += Gr[i][k] * Ui[k][j] + Gi[i][k] * Ur[k][j];
      }
      Rr[i][j] = ar; Ri[i][j] = ai;
    }
  for (int i = 0; i < 4; ++i)
    for (int j = 0; j < 4; ++j) { Ur[i][j] = Rr[i][j]; Ui[i][j] = Ri[i][j]; }
}

// Expand 2x2 gate (gr,gi) to 4x4 on local pair basis i = 2*b_q1 + b_q2 and
// left-multiply into U.  on_low: gate acts on q2 (low local bit) else q1.
// ctrl: if nonzero, gate applies only when the *other* bit is 1 (identity when 0).
__device__ void apply_gate(float Ur[4][4], float Ui[4][4],
                           const float gr[2][2], const float gi[2][2],
                           int on_low, int ctrl) {
  float Gr[4][4] = {}, Gi[4][4] = {};
  for (int a = 0; a < 4; ++a)
    for (int b = 0; b < 4; ++b) {
      int a1 = a >> 1, a2 = a & 1, b1 = b >> 1, b2 = b & 1;
      int at = on_low ? a2 : a1, bt = on_low ? b2 : b1;   // target-qubit bits
      int ao = on_low ? a1 : a2, bo = on_low ? b1 : b2;   // other-qubit bits
      if (ao != bo) continue;
      if (ctrl && ao == 0) { if (at == bt) Gr[a][b] = 1.f; continue; }
      Gr[a][b] = gr[at][bt];
      Gi[a][b] = gi[at][bt];
    }
  cmul4(Gr, Gi, Ur, Ui);
}

__global__ void pair_block_kernel(const float* __restrict__ params,
                                  float* __restrict__ ublk) {
  int k = threadIdx.x;
  if (k >= NPAIRS) return;
  const float* p = params + k * 10;

  float Ur[4][4] = {}, Ui[4][4] = {};
  Ur[0][0] = Ur[1][1] = Ur[2][2] = Ur[3][3] = 1.f;

  float gr[2][2], gi[2][2];
  auto RX = [&](float th) {            // [[c, -i s],[-i s, c]]
    float h = 0.5f * th, c = cosf(h), s = sinf(h);
    gr[0][0] = c;  gr[0][1] = 0;  gr[1][0] = 0;  gr[1][1] = c;
    gi[0][0] = 0;  gi[0][1] = -s; gi[1][0] = -s; gi[1][1] = 0;
  };
  auto RZ = [&](float th) {            // diag(e^{-ih}, e^{+ih})
    float h = 0.5f * th, c = cosf(h), s = sinf(h);
    gr[0][0] = c;  gr[0][1] = 0;  gr[1][0] = 0;  gr[1][1] = c;
    gi[0][0] = -s; gi[0][1] = 0;  gi[1][0] = 0;  gi[1][1] = s;
  };
  auto RY = [&](float th) {            // real rotation
    float h = 0.5f * th, c = cosf(h), s = sinf(h);
    gr[0][0] = c;  gr[0][1] = -s; gr[1][0] = s;  gr[1][1] = c;
    gi[0][0] = 0;  gi[0][1] = 0;  gi[1][0] = 0;  gi[1][1] = 0;
  };

  RX(p[0]); apply_gate(Ur, Ui, gr, gi, 0, 0);
  RZ(p[1]); apply_gate(Ur, Ui, gr, gi, 0, 0);
  RX(p[2]); apply_gate(Ur, Ui, gr, gi, 1, 0);
  RZ(p[3]); apply_gate(Ur, Ui, gr, gi, 1, 0);
  RY(p[4]); apply_gate(Ur, Ui, gr, gi, 1, 1);  // CRY ctrl=q1, tgt=q2
  RX(p[5]); apply_gate(Ur, Ui, gr, gi, 0, 0);
  RZ(p[6]); apply_gate(Ur, Ui, gr, gi, 0, 0);
  RX(p[7]); apply_gate(Ur, Ui, gr, gi, 1, 0);
  RZ(p[8]); apply_gate(Ur, Ui, gr, gi, 1, 0);
  RY(p[9]); apply_gate(Ur, Ui, gr, gi, 0, 1);  // CRY ctrl=q2, tgt=q1

  float* o = ublk + k * 32;
  for (int r = 0; r < 4; ++r)
    for (int c = 0; c < 4; ++c) { o[r * 4 + c] = Ur[r][c]; o[16 + r * 4 + c] = Ui[r][c]; }
}

// ============================================================================
// Kernel 2: one workgroup per batch sample; statevector resident in LDS.
// Each pair-block = 16x16x16 real GEMM over groups via V_WMMA_F32_16X16X4_F32.
// ============================================================================

// A = blockdiag(M8, M8), M8 = [[Ur,-Ui],[Ui,Ur]]; ub = 32 floats {Ur, Ui}.
__device__ __forceinline__ float a_elem(const float* ub, int m, int k) {
  bool on = (m < 8) == (k < 8);
  int r = m & 7, c = k & 7;
  float ur = ub[(r & 3) * 4 + (c & 3)];
  float ui = ub[16 + (r & 3) * 4 + (c & 3)];
  float v = (r < 4) ? ((c < 4) ? ur : -ui) : ((c < 4) ? ui : ur);
  return on ? v : 0.f;
}

__global__ __launch_bounds__(THREADS)
void qsim_kernel(const float* __restrict__ inputs,
                 const float* __restrict__ ublk,
                 float* __restrict__ out) {
  extern __shared__ float smem[];
  float* sre  = smem;                   // [NSTATE] real parts
  float* sim_ = smem + NSTATE;          // [NSTATE] imag parts
  float* cq   = smem + 2 * NSTATE;      // [WIRES] cos(h_q)
  float* sq   = cq + WIRES;             // [WIRES] sin(h_q)
  float* zacc = sq + WIRES;             // [WIRES] <Z_q> accumulators
  float* ub   = zacc + WIRES;           // [32] current block Ur/Ui

  const int tid  = threadIdx.x;
  const int samp = blockIdx.x;
  const int lane = tid & 31;
  const int half = lane >> 4;           // half-wave select
  const int col  = lane & 15;           // M (A/D) or N (B/D) index
  const int wave = tid >> 5;            // 0..7

  if (tid < WIRES) {
    float h = 0.5f * inputs[samp * WIRES + tid];
    cq[tid] = cosf(h);
    sq[tid] = sinf(h);
    zacc[tid] = 0.f;
  }
  __syncthreads();

  // ---- product-state init (RY encoding applied to |0...0> is separable) ----
  // flat-index bit of qubit q is bit (13-q); thread owns idx = tid*64 + j.
  float ph = 1.f;
#pragma unroll
  for (int q = 0; q < 8; ++q) ph *= ((tid >> (7 - q)) & 1) ? sq[q] : cq[q];
  for (int j = 0; j < APT; ++j) {
    float pp = ph;
#pragma unroll
    for (int q = 8; q < WIRES; ++q) pp *= ((j >> (13 - q)) & 1) ? sq[q] : cq[q];
    int idx = (tid << 6) | j;
    sre[idx] = pp;
    sim_[idx] = 0.f;
  }

  // ---- variational pair blocks ----
  for (int k = 0; k < NPAIRS; ++k) {
    __syncthreads();                           // prior block's stores visible
    if (tid < 32) ub[tid] = ublk[k * 32 + tid];
    __syncthreads();

    const int q1   = (k < 7) ? (2 * k) : (2 * (k - 7) + 1);
    const int p2   = 12 - q1;                  // bit position of qubit q1+1
    const int mlow = (1 << p2) - 1;

    // A fragments: 4 ops of 16x16x4; lane holds A(m=col, k=4t+2*half+j).
    v2f afrag[4];
#pragma unroll
    for (int t = 0; t < 4; ++t) {
#pragma unroll
      for (int j = 0; j < 2; ++j)
        afrag[t][j] = a_elem(ub, col, 4 * t + 2 * half + j);
    }

    // 4096 groups / (32 groups * 8 waves) = 16 iterations per wave.
    for (int it = 0; it < 16; ++it) {
      int gbase = (wave * 16 + it) * 32;
      int ga = gbase + col;                    // group set 0 (K/M rows 0..7)
      int gb = gbase + 16 + col;               // group set 1 (K/M rows 8..15)
      int ia = ((ga >> p2) << (p2 + 2)) | (ga & mlow);
      int ib = ((gb >> p2) << (p2 + 2)) | (gb & mlow);

      v8f d = {0.f, 0.f, 0.f, 0.f, 0.f, 0.f, 0.f, 0.f};
#pragma unroll
      for (int t = 0; t < 4; ++t) {
        v2f b;
#pragma unroll
        for (int j = 0; j < 2; ++j) {
          int kk   = 4 * t + 2 * half + j;     // B row: lane=N, (half,vgpr)=K
          int comp = kk & 7;                   // 0..3 re, 4..7 im of amp comp&3
          int base = (kk < 8) ? ia : ib;
          int addr = base + ((comp & 3) << p2);
          b[j] = (comp < 4) ? sre[addr] : sim_[addr];
        }
        d = __builtin_amdgcn_wmma_f32_16x16x4_f32(
                false, afrag[t], false, b, (short)0, d, false, false);
      }

      // D: vgpr r, lanes 0-15 = row r (set 0), lanes 16-31 = row r+8 (set 1).
      int ibase = half ? ib : ia;
#pragma unroll
      for (int r = 0; r < 4; ++r) sre[ibase + (r << p2)] = d[r];
#pragma unroll
      for (int r = 0; r < 4; ++r) sim_[ibase + (r << p2)] = d[r + 4];
    }
  }
  __syncthreads();

  // ---- <Z_q> readout: sum of signed probabilities ----
  float acc[WIRES];
#pragma unroll
  for (int q = 0; q < WIRES; ++q) acc[q] = 0.f;
  for (int j = 0; j < APT; ++j) {
    int idx = (tid << 6) | j;
    float re = sre[idx], im = sim_[idx];
    float p = re * re + im * im;
#pragma unroll
    for (int q = 0; q < WIRES; ++q)
      acc[q] += ((idx >> (13 - q)) & 1) ? -p : p;
  }
#pragma unroll
  for (int q = 0; q < WIRES; ++q) atomicAdd(&zacc[q], acc[q]);
  __syncthreads();
  if (tid < WIRES) out[samp * WIRES + tid] = zacc[tid];
}

// ============================================================================

extern "C" void kernel_launch(void* const* d_in, const int* in_sizes, int n_in,
                              void* d_out, int out_size, void* d_ws, size_t ws_size,
                              hipStream_t stream) {
  const float* inputs = (const float*)d_in[0];   // (BATCH, 14) float32
  const float* params = (const float*)d_in[1];   // (13, 10)  float32
  float* out  = (float*)d_out;                   // (BATCH, 14) float32
  float* ublk = (float*)d_ws;                    // 13 * 32 floats

  int batch = in_sizes[0] / WIRES;

  pair_block_kernel<<<1, 32, 0, stream>>>(params, ublk);

  size_t shmem = (size_t)(2 * NSTATE + 3 * WIRES + 32) * sizeof(float);
  qsim_kernel<<<batch, THREADS, shmem, stream>>>(inputs, ublk, out);
}